// Sampler_53455162966582
// MI455X (gfx1250) — compile-verified
//
#include <hip/hip_runtime.h>
#include <stdint.h>

// ---------------------------------------------------------------------------
// Segmented-softmax sampler for MI455X (gfx1250, wave32).
// Memory-bound: ~300-450 MB total traffic -> ~15-20us at 23.3 TB/s.
// No matrix math exists in this op; the win is pure data movement:
//   - b128 vector streams, global_prefetch_b8 ahead of the stream
//   - temporal hints: single-use streams NT so the 192MB L2 keeps the
//     reused buffers (prob table, seg ids, e-cache) resident
//   - wave32-uniform segment detection + shfl_xor reduction -> 1 atomic/wave
//   - order-independent atomics only (u32 max, u64 fixed-point add)
//     so graph replays are bit-deterministic.
// ---------------------------------------------------------------------------

#define N_EG_SEGS 4096

typedef int   i4 __attribute__((ext_vector_type(4)));
typedef float f4 __attribute__((ext_vector_type(4)));

static constexpr float kTemp    = 1.0f;
static constexpr float kFixF    = 274877906944.0f;      // 2^38
static constexpr float kInvFixF = 1.0f / 274877906944.0f;

// Order-preserving map: float -> uint so unsigned atomicMax == float max.
__device__ __forceinline__ unsigned encOrder(float f) {
    unsigned u = __float_as_uint(f);
    return (u & 0x80000000u) ? ~u : (u | 0x80000000u);
}
__device__ __forceinline__ float decOrder(unsigned u) {
    unsigned b = (u & 0x80000000u) ? (u & 0x7fffffffu) : ~u;
    return __uint_as_float(b);
}

__global__ void k_init(unsigned* __restrict__ smax,
                       unsigned long long* __restrict__ ssum) {
    int i = blockIdx.x * blockDim.x + threadIdx.x;
    if (i < N_EG_SEGS) { smax[i] = 0u; ssum[i] = 0ull; }
}

// Pass 1: logits = prob[eid] + llu ; segment max (optionally store logits).
// STORE==true: eid/llu are never read again -> non-temporal loads.
template <bool STORE>
__global__ void k_logit_max(const float* __restrict__ prob,
                            const float* __restrict__ llu,
                            const int*   __restrict__ eid,
                            const int*   __restrict__ sid,
                            unsigned*    __restrict__ smax,
                            float*       __restrict__ lbuf,
                            int n) {
    const int lane     = threadIdx.x & 31;
    const int tid      = blockIdx.x * blockDim.x + threadIdx.x;
    const int nthreads = gridDim.x * blockDim.x;
    const int n4       = (n + 3) >> 2;

    for (int t = tid; ; t += nthreads) {
        bool inrange = (t < n4);
        if (!__any(inrange)) break;          // wave-uniform exit: no partial-EXEC shuffles

        bool  full = false;
        int   seg  = -1;
        float m    = -__builtin_inff();
        float l0 = 0.f, l1 = 0.f, l2 = 0.f, l3 = 0.f;
        i4    s4 = {-1, -1, -1, -1};

        if (inrange) {
            int i0 = t << 2;
            if (i0 + 3 < n) {
                i4 e4;
                f4 u4;
                if (STORE) {   // single-use streams: NT keeps L2 for prob/sid/lbuf
                    e4 = __builtin_nontemporal_load((const i4*)eid + t);
                    u4 = __builtin_nontemporal_load((const f4*)llu + t);
                } else {       // re-read in pass 2: regular temporal
                    e4 = ((const i4*)eid)[t];
                    u4 = ((const f4*)llu)[t];
                }
                s4 = ((const i4*)sid)[t];    // reused in passes 2-3: RT
                l0 = prob[e4.x] + u4.x;
                l1 = prob[e4.y] + u4.y;
                l2 = prob[e4.z] + u4.z;
                l3 = prob[e4.w] + u4.w;
                if (STORE) {
                    f4 lv = {l0, l1, l2, l3};
                    ((f4*)lbuf)[t] = lv;     // reread pass 2: RT store
                }
                full = true;
            } else {
                for (int k = 0; k < 4; ++k) {
                    int i = i0 + k;
                    if (i < n) {
                        float l = prob[eid[i]] + llu[i];
                        if (STORE) lbuf[i] = l;
                        atomicMax(&smax[sid[i]], encOrder(l));
                    }
                }
            }
            // gfx1250 global_prefetch_b8 for next grid-stride iteration
            size_t nx = ((size_t)(t + nthreads)) << 2;
            __builtin_prefetch(eid + nx, 0, 3);
            __builtin_prefetch(llu + nx, 0, 3);
            __builtin_prefetch(sid + nx, 0, 3);
        }

        bool same4 = full && (s4.x == s4.y) && (s4.x == s4.z) && (s4.x == s4.w);
        if (same4) { seg = s4.x; m = fmaxf(fmaxf(l0, l1), fmaxf(l2, l3)); }

        int seg0 = __shfl(seg, 0, 32);
        if (__all(same4 && (seg == seg0))) {
            // whole wave on one segment: butterfly reduce, one atomic
            #pragma unroll
            for (int off = 16; off > 0; off >>= 1)
                m = fmaxf(m, __shfl_xor(m, off, 32));
            if (lane == 0) atomicMax(&smax[seg0], encOrder(m));
        } else if (full) {
            if (same4) {
                atomicMax(&smax[seg], encOrder(m));
            } else {
                atomicMax(&smax[s4.x], encOrder(l0));
                atomicMax(&smax[s4.y], encOrder(l1));
                atomicMax(&smax[s4.z], encOrder(l2));
                atomicMax(&smax[s4.w], encOrder(l3));
            }
        }
    }
}

// Pass 2: e = exp(logit - segmax); fixed-point u64 segment sum (deterministic).
// If STORED, logits are read from lbuf and overwritten with e for pass 3.
// If !STORED, this is the last read of eid/llu -> non-temporal.
template <bool STORED>
__global__ void k_exp_sum(const float* __restrict__ prob,
                          const float* __restrict__ llu,
                          const int*   __restrict__ eid,
                          const int*   __restrict__ sid,
                          const unsigned* __restrict__ smax,
                          unsigned long long* __restrict__ ssum,
                          float* __restrict__ lbuf,
                          int n) {
    const int lane     = threadIdx.x & 31;
    const int tid      = blockIdx.x * blockDim.x + threadIdx.x;
    const int nthreads = gridDim.x * blockDim.x;
    const int n4       = (n + 3) >> 2;

    for (int t = tid; ; t += nthreads) {
        bool inrange = (t < n4);
        if (!__any(inrange)) break;

        bool full = false;
        int  seg  = -1;
        unsigned long long q = 0ull;
        i4 s4 = {-1, -1, -1, -1};
        float e0 = 0.f, e1 = 0.f, e2 = 0.f, e3 = 0.f;

        if (inrange) {
            int i0 = t << 2;
            if (i0 + 3 < n) {
                s4 = ((const i4*)sid)[t];
                float l0, l1, l2, l3;
                if (STORED) {
                    f4 lv = ((const f4*)lbuf)[t];
                    l0 = lv.x; l1 = lv.y; l2 = lv.z; l3 = lv.w;
                } else {
                    i4 e4 = __builtin_nontemporal_load((const i4*)eid + t);
                    f4 u4 = __builtin_nontemporal_load((const f4*)llu + t);
                    l0 = prob[e4.x] + u4.x;
                    l1 = prob[e4.y] + u4.y;
                    l2 = prob[e4.z] + u4.z;
                    l3 = prob[e4.w] + u4.w;
                }
                e0 = __expf(l0 - decOrder(smax[s4.x]));
                e1 = __expf(l1 - decOrder(smax[s4.y]));
                e2 = __expf(l2 - decOrder(smax[s4.z]));
                e3 = __expf(l3 - decOrder(smax[s4.w]));
                if (STORED) {
                    f4 ev = {e0, e1, e2, e3};
                    ((f4*)lbuf)[t] = ev;     // gathered in pass 3: RT store
                }
                full = true;
            } else {
                for (int k = 0; k < 4; ++k) {
                    int i = i0 + k;
                    if (i < n) {
                        int   s = sid[i];
                        float l = STORED ? lbuf[i] : (prob[eid[i]] + llu[i]);
                        float e = __expf(l - decOrder(smax[s]));
                        if (STORED) lbuf[i] = e;
                        atomicAdd(&ssum[s],
                                  (unsigned long long)(e * kFixF + 0.5f));
                    }
                }
            }
            size_t nx = ((size_t)(t + nthreads)) << 2;
            __builtin_prefetch(sid + nx, 0, 3);
            if (STORED) {
                __builtin_prefetch(lbuf + nx, 0, 3);
            } else {
                __builtin_prefetch(eid + nx, 0, 3);
                __builtin_prefetch(llu + nx, 0, 3);
            }
        }

        bool same4 = full && (s4.x == s4.y) && (s4.x == s4.z) && (s4.x == s4.w);
        if (same4) {
            seg = s4.x;
            q = (unsigned long long)(e0 * kFixF + 0.5f)
              + (unsigned long long)(e1 * kFixF + 0.5f)
              + (unsigned long long)(e2 * kFixF + 0.5f)
              + (unsigned long long)(e3 * kFixF + 0.5f);
        }

        int seg0 = __shfl(seg, 0, 32);
        if (__all(same4 && (seg == seg0))) {
            #pragma unroll
            for (int off = 16; off > 0; off >>= 1)
                q += __shfl_xor(q, off, 32);
            if (lane == 0) atomicAdd(&ssum[seg0], q);
        } else if (full) {
            if (same4) {
                atomicAdd(&ssum[seg], q);
            } else {
                atomicAdd(&ssum[s4.x], (unsigned long long)(e0 * kFixF + 0.5f));
                atomicAdd(&ssum[s4.y], (unsigned long long)(e1 * kFixF + 0.5f));
                atomicAdd(&ssum[s4.z], (unsigned long long)(e2 * kFixF + 0.5f));
                atomicAdd(&ssum[s4.w], (unsigned long long)(e3 * kFixF + 0.5f));
            }
        }
    }
}

// Pass 3: gather sampled candidates, p = e / segsum, straight-through output.
template <bool STORED>
__global__ void k_gather_out(const float* __restrict__ prob,
                             const float* __restrict__ llu,
                             const int*   __restrict__ eid,
                             const int*   __restrict__ sid,
                             const int*   __restrict__ cai,
                             const unsigned* __restrict__ smax,
                             const unsigned long long* __restrict__ ssum,
                             const float* __restrict__ lbuf,
                             float* __restrict__ out,
                             int n_samp) {
    const int tid      = blockIdx.x * blockDim.x + threadIdx.x;
    const int nthreads = gridDim.x * blockDim.x;
    for (int j = tid; j < n_samp; j += nthreads) {
        int c = __builtin_nontemporal_load(cai + j);   // single-use stream
        int s = sid[c];
        float e;
        if (STORED) {
            e = lbuf[c];                               // random gather, L2-resident
        } else {
            float l = prob[eid[c]] + llu[c];
            e = __expf(l - decOrder(smax[s]));
        }
        float denom = (float)ssum[s] * kInvFixF;
        float p = e / denom;
        float y = p / kTemp;
        float r = (1.0f - y) + y;   // replicate reference float ordering
        __builtin_nontemporal_store(r, out + j);       // write-once output
        __builtin_prefetch(cai + (size_t)(j + nthreads), 0, 3);
    }
}

extern "C" void kernel_launch(void* const* d_in, const int* in_sizes, int n_in,
                              void* d_out, int out_size, void* d_ws, size_t ws_size,
                              hipStream_t stream) {
    const float* prob = (const float*)d_in[0];   // prob_params [N_FULL_EDGES]
    const float* llu  = (const float*)d_in[1];   // loglog_u    [N_CAND]
    const int*   eid  = (const int*)d_in[2];     // edge_id     [N_CAND]
    const int*   sid  = (const int*)d_in[3];     // seg_ids     [N_CAND] (sorted)
    const int*   cai  = (const int*)d_in[4];     // ca_idx      [N_SAMP]

    const int n_cand = in_sizes[1];
    const int n_samp = in_sizes[4];

    // ws layout: [0,16K) u32 segmax | [16K,48K) u64 segsum | [64K, ...) e-cache
    unsigned*           smax = (unsigned*)d_ws;
    unsigned long long* ssum = (unsigned long long*)((char*)d_ws + 16384);
    float*              lbuf = (float*)((char*)d_ws + 65536);
    const bool stored = ws_size >= (size_t)65536 + (size_t)n_cand * sizeof(float);

    float* out = (float*)d_out;

    const int threads = 256;
    k_init<<<(N_EG_SEGS + threads - 1) / threads, threads, 0, stream>>>(smax, ssum);

    const int n4 = (n_cand + 3) / 4;
    int blocks = (n4 + threads - 1) / threads;
    if (blocks > 2048) blocks = 2048;            // grid-stride, keeps prefetch useful
    int blocks3 = (n_samp + threads - 1) / threads;
    if (blocks3 > 2048) blocks3 = 2048;

    if (stored) {
        k_logit_max<true><<<blocks, threads, 0, stream>>>(prob, llu, eid, sid, smax, lbuf, n_cand);
        k_exp_sum<true><<<blocks, threads, 0, stream>>>(prob, llu, eid, sid, smax, ssum, lbuf, n_cand);
        k_gather_out<true><<<blocks3, threads, 0, stream>>>(prob, llu, eid, sid, cai, smax, ssum, lbuf, out, n_samp);
    } else {
        k_logit_max<false><<<blocks, threads, 0, stream>>>(prob, llu, eid, sid, smax, lbuf, n_cand);
        k_exp_sum<false><<<blocks, threads, 0, stream>>>(prob, llu, eid, sid, smax, ssum, lbuf, n_cand);
        k_gather_out<false><<<blocks3, threads, 0, stream>>>(prob, llu, eid, sid, cai, smax, ssum, lbuf, out, n_samp);
    }
}